// VoxelSampler_40381282517052
// MI455X (gfx1250) — compile-verified
//
#include <hip/hip_runtime.h>
#include <math.h>

#define GAMMA 1.1f

typedef __attribute__((ext_vector_type(2))) float v2f;
typedef __attribute__((ext_vector_type(8))) float v8f;

// ---------------------------------------------------------------------------
// Kernel 1: zero the output, build per-box data:
//   boxdata[m] = (-2*bx, -2*by, bx^2+by^2, 0)   (A-matrix rows for WMMA)
//   rad2[m]    = (||(dx,dy)*0.5|| * GAMMA)^2
// ---------------------------------------------------------------------------
__global__ void prep_kernel(const float* __restrict__ boxes, int M,
                            float* __restrict__ out, int out_n,
                            float4* __restrict__ boxdata,
                            float* __restrict__ rad2) {
  int t = blockIdx.x * blockDim.x + threadIdx.x;
  if (t < out_n) out[t] = 0.0f;
  if (t < M) {
    float bx = boxes[t * 9 + 0], by = boxes[t * 9 + 1];
    float dx = boxes[t * 9 + 3], dy = boxes[t * 9 + 4];
    float hx = 0.5f * dx, hy = 0.5f * dy;
    float r = sqrtf(hx * hx + hy * hy) * GAMMA;
    boxdata[t] = make_float4(-2.0f * bx, -2.0f * by, bx * bx + by * by, 0.0f);
    rad2[t] = r * r;
  }
}

// ---------------------------------------------------------------------------
// Kernel 2: WMMA pairwise-distance + mask/count.
// One wave per (box-tile of 16, chunk of 1024 points).
// d^2 = A(16x4) x B(4x16): A row m = [-2bx, -2by, 1, cmag],
//                          B col n = [ px ,  py , pp, 1  ],  pp = px^2+py^2.
// A layout: lanes0-15 hold K0,K1 (VGPR0,1); lanes16-31 hold K2,K3.
// D layout: VGPR r -> box r (lanes0-15) / box r+8 (lanes16-31), lane = point.
// ballot(d2<=r2) gives {hi16: box r+8, lo16: box r} point masks.
// Unrolled x2: two independent B operands / WMMA chains in flight per step.
// ---------------------------------------------------------------------------
__global__ __launch_bounds__(32)
void count_kernel(const float* __restrict__ P, int N,
                  const float4* __restrict__ boxdata,
                  const float* __restrict__ rad2,
                  unsigned* __restrict__ masks, int ngroup,
                  int* __restrict__ cnt, int nch) {
  const int wave = blockIdx.x;
  const int bt   = wave / nch;   // box tile (16 boxes)
  const int c    = wave % nch;   // 1024-point chunk
  const int lane = threadIdx.x;
  const bool lo  = lane < 16;

  // A operand (boxes), built once.
  float4 bd = boxdata[bt * 16 + (lane & 15)];
  v2f a;
  a.x = lo ? bd.x : 1.0f;  // K0 : K2
  a.y = lo ? bd.y : bd.z;  // K1 : K3

  // Per-lane r^2 matching the D layout (lanes 0-15: box r, 16-31: box r+8).
  float r2v[8];
#pragma unroll
  for (int r = 0; r < 8; ++r) r2v[r] = rad2[bt * 16 + r + (lo ? 0 : 8)];

  int cl[8], ch[8];
#pragma unroll
  for (int r = 0; r < 8; ++r) { cl[r] = 0; ch[r] = 0; }

  const int g0 = c * 64;
  for (int k = 0; k < 64; k += 2) {
    int idx0 = (g0 + k) * 16 + (lane & 15);
    int idx1 = idx0 + 16;
    float px0, py0, px1, py1;
    if (idx0 < N) { px0 = P[(size_t)idx0 * 5]; py0 = P[(size_t)idx0 * 5 + 1]; }
    else          { px0 = 1e18f; py0 = 1e18f; }
    if (idx1 < N) { px1 = P[(size_t)idx1 * 5]; py1 = P[(size_t)idx1 * 5 + 1]; }
    else          { px1 = 1e18f; py1 = 1e18f; }

    // Near (WGP-scope) prefetch two groups ahead; wraps so it's always
    // in-range (no loop peel).
    {
      int pk = (k + 2) & 63;
      int nidx = (g0 + pk) * 16 + (lane & 15);
      if (nidx < N) __builtin_prefetch(&P[(size_t)nidx * 5], 0, 3);
    }

    float pp0 = px0 * px0 + py0 * py0;
    float pp1 = px1 * px1 + py1 * py1;
    v2f b0, b1;
    b0.x = lo ? px0 : pp0;  b0.y = lo ? py0 : 1.0f;
    b1.x = lo ? px1 : pp1;  b1.y = lo ? py1 : 1.0f;

    v8f cz = {0.f, 0.f, 0.f, 0.f, 0.f, 0.f, 0.f, 0.f};
    v8f d0 = __builtin_amdgcn_wmma_f32_16x16x4_f32(false, a, false, b0,
                                                   (short)0, cz, false, false);
    v8f d1 = __builtin_amdgcn_wmma_f32_16x16x4_f32(false, a, false, b1,
                                                   (short)0, cz, false, false);
    unsigned w0 = 0, w1 = 0;
#pragma unroll
    for (int r = 0; r < 8; ++r) {
      unsigned bal0 = __builtin_amdgcn_ballot_w32(d0[r] <= r2v[r]);
      unsigned bal1 = __builtin_amdgcn_ballot_w32(d1[r] <= r2v[r]);
      cl[r] += __popc(bal0 & 0xFFFFu) + __popc(bal1 & 0xFFFFu);
      ch[r] += __popc(bal0 >> 16)     + __popc(bal1 >> 16);
      w0 = (lane == r) ? bal0 : w0;   // lane r keeps ballot r
      w1 = (lane == r) ? bal1 : w1;
    }
    if (lane < 8) {
      unsigned* mp = &masks[((size_t)bt * ngroup + (g0 + k)) * 8 + lane];
      mp[0] = w0;   // group k
      mp[8] = w1;   // group k+1
    }
  }

  if (lane == 0) {
#pragma unroll
    for (int r = 0; r < 8; ++r) {
      cnt[(bt * 16 + r)     * nch + c] = cl[r];
      cnt[(bt * 16 + r + 8) * nch + c] = ch[r];
    }
  }
}

// ---------------------------------------------------------------------------
// Kernel 3: per-box exclusive prefix sum over chunk counts (blockDim == nch).
// ---------------------------------------------------------------------------
__global__ void scan_kernel(const int* __restrict__ cnt,
                            int* __restrict__ baseArr, int nch) {
  __shared__ int s[1024];
  int m = blockIdx.x;
  int t = threadIdx.x;
  s[t] = cnt[(size_t)m * nch + t];
  __syncthreads();
  for (int off = 1; off < nch; off <<= 1) {
    int x = s[t];
    int y = (t >= off) ? s[t - off] : 0;
    __syncthreads();
    s[t] = x + y;
    __syncthreads();
  }
  baseArr[(size_t)m * nch + t] = (t == 0) ? 0 : s[t - 1];
}

// ---------------------------------------------------------------------------
// Kernel 4: ordered compaction + gather. One wave per (box, chunk); chunks
// whose starting slot >= S exit immediately. Two 16-point groups per step
// (lanes 0-15 -> group k, 16-31 -> group k+1), ballot-prefix gives the slot.
// ---------------------------------------------------------------------------
__global__ __launch_bounds__(32)
void emit_kernel(const float* __restrict__ P,
                 const float* __restrict__ boxes,
                 const unsigned* __restrict__ masks, int ngroup,
                 const int* __restrict__ baseArr, int nch,
                 float* __restrict__ out, int S) {
  const int blk = blockIdx.x;
  const int m = blk / nch;
  const int c = blk % nch;
  int run = baseArr[(size_t)m * nch + c];
  if (run >= S) return;

  const int lane = threadIdx.x;
  const int bt = m >> 4, j = m & 15;
  const int ridx  = j & 7;
  const int shift = ((j >> 3) << 4) + (lane & 15);
  const float vx = boxes[m * 9 + 7];
  const float vy = boxes[m * 9 + 8];
  const unsigned ltmask = (1u << lane) - 1u;
  const int g0 = c * 64;

  for (int k = 0; k < 64; k += 2) {
    int G = g0 + k + (lane >> 4);
    unsigned w = masks[((size_t)bt * ngroup + G) * 8 + ridx];
    int bit = (int)((w >> shift) & 1u);
    unsigned bal = __builtin_amdgcn_ballot_w32(bit != 0);
    int slot = run + __popc(bal & ltmask);
    if (bit && slot < S) {
      int pidx = G * 16 + (lane & 15);
      const float* ps = P + (size_t)pidx * 5;
      float* o = out + ((size_t)m * S + slot) * 7;
      o[0] = ps[0]; o[1] = ps[1]; o[2] = ps[2];
      o[3] = ps[3]; o[4] = ps[4];
      o[5] = vx;    o[6] = vy;
    }
    run += __popc(bal);
    if (run >= S) break;
  }
}

// ---------------------------------------------------------------------------
extern "C" void kernel_launch(void* const* d_in, const int* in_sizes, int n_in,
                              void* d_out, int out_size, void* d_ws, size_t ws_size,
                              hipStream_t stream) {
  const float* P = (const float*)d_in[0];   // [N,5]
  const float* B = (const float*)d_in[1];   // [M,9]
  float* out = (float*)d_out;               // [M,S,7]

  const int N = in_sizes[0] / 5;            // 262144
  const int M = in_sizes[1] / 9;            // 256
  const int S = out_size / (M * 7);         // 128
  const int nch    = (N + 1023) / 1024;     // 256 chunks of 1024 points
  const int ngroup = nch * 64;              // 16-point groups
  const int bt     = M / 16;                // box tiles

  // Workspace layout (all 16B-aligned): masks | cnt | base | boxdata | rad2
  char* ws = (char*)d_ws;
  unsigned* masks   = (unsigned*)ws;
  size_t mbytes     = (size_t)bt * ngroup * 8 * sizeof(unsigned);   // 8 MB
  int* cnt          = (int*)(ws + mbytes);
  int* baseA        = cnt + (size_t)M * nch;
  float4* boxdata   = (float4*)(baseA + (size_t)M * nch);
  float* rad2       = (float*)(boxdata + M);

  prep_kernel<<<(out_size + 255) / 256, 256, 0, stream>>>(B, M, out, out_size,
                                                          boxdata, rad2);
  count_kernel<<<bt * nch, 32, 0, stream>>>(P, N, boxdata, rad2,
                                            masks, ngroup, cnt, nch);
  scan_kernel<<<M, nch, 0, stream>>>(cnt, baseA, nch);
  emit_kernel<<<M * nch, 32, 0, stream>>>(P, B, masks, ngroup, baseA, nch,
                                          out, S);
}